// Backward_projection_73598559584915
// MI455X (gfx1250) — compile-verified
//
#include <hip/hip_runtime.h>
#include <stdint.h>

typedef __attribute__((ext_vector_type(2))) float v2f;
typedef __attribute__((ext_vector_type(8))) float v8f;
typedef __attribute__((ext_vector_type(4))) unsigned int vu4;
typedef __attribute__((ext_vector_type(4))) int vi4;
typedef __attribute__((ext_vector_type(8))) int vi8;

#define NN    128
#define AA    285
#define DD    183
#define BATCH 256
#define KP    184              // K padded to multiple of 4
#define NP    192              // N padded to multiple of 16
#define MM    (BATCH * AA)     // 72960 rows (B*A)
#define TPB   4                // M tiles per block (persistent, double-buffered)

#define PI_D 3.14159265358979323846

// ---------------------------------------------------------------------------
// Kernel 1: build padded ramp-filter matrix Hp[KP][NP],
//   Hp[k][j] = 12*DC*h(j-k), zero outside [0,183)x[0,183)
// ---------------------------------------------------------------------------
__global__ void __launch_bounds__(256)
build_h_kernel(float* __restrict__ Hp) {
  int idx = blockIdx.x * blockDim.x + threadIdx.x;
  if (idx >= KP * NP) return;
  int k = idx / NP;
  int j = idx % NP;
  double val = 0.0;
  if (k < DD && j < DD) {
    const double RHO = 28.284271247461902;   // sqrt(2)*20
    const double DC  = 2.0 * RHO / (double)DD;
    int m = j - k;
    if (m == 0) {
      val = 12.0 * DC / (4.0 * DC * DC);     // 3/DC
    } else if (m & 1) {                      // odd (works for negatives too)
      double pm = PI_D * (double)m * DC;
      val = -12.0 * DC / (pm * pm);
    }
  }
  Hp[idx] = (float)val;
}

// ---------------------------------------------------------------------------
// TDM: DMA one 16x184 f32 tile of x into LDS.
// D# per CDNA5 ISA ch.8:
//   group0: count=1 | lds_addr | global_addr(57b) | type=2
//   group1: data_size=2 (4B), tensor_dim0=183 (OOB-zero fills column 183),
//           tensor_dim1=16, tile_dim0=184, tile_dim1=16, dim0_stride=183
// ---------------------------------------------------------------------------
__device__ __forceinline__ void tdm_load_tile(unsigned lds_addr,
                                              const float* gptr) {
  unsigned long long ga = (unsigned long long)(uintptr_t)gptr;
  vu4 g0;
  g0[0] = 1u;                                        // count=1, user mode
  g0[1] = lds_addr;                                  // lds_addr (bytes)
  g0[2] = (unsigned)(ga & 0xFFFFFFFFu);              // global_addr[31:0]
  g0[3] = (unsigned)((ga >> 32) & 0x01FFFFFFu)       // global_addr[56:32]
          | (2u << 30);                              // type=2 ("image")
  vi8 g1;
  g1[0] = 0x00020000;                                // data_size=2 (4 bytes)
  g1[1] = (int)(183u << 16);                         // tensor_dim0 = 183
  g1[2] = (int)(16u << 16);                          // tensor_dim1 = 16
  g1[3] = (int)(184u << 16);                         // tile_dim0   = 184
  g1[4] = 16;                                        // tile_dim1   = 16
  g1[5] = 183;                                       // tensor_dim0_stride
  g1[6] = 0;
  g1[7] = 0;
  vi4 z4 = {0, 0, 0, 0};
#if __clang_major__ >= 23
  vi8 z8 = {0, 0, 0, 0, 0, 0, 0, 0};
  __builtin_amdgcn_tensor_load_to_lds(g0, g1, z4, z4, z8, 0);
#else
  __builtin_amdgcn_tensor_load_to_lds(g0, g1, z4, z4, 0);
#endif
}

// ---------------------------------------------------------------------------
// Kernel 2: ramp filtering as f32 WMMA GEMM.
//   q[r][j] = sum_k x[r][k] * Hp[k][j]   (r in [0,72960), j in [0,192))
// Block = 384 threads = 12 waves; block owns TPB=4 16-row M tiles,
// TDM double-buffers x tiles into LDS while waves run 16x16x4 f32 WMMA.
// ---------------------------------------------------------------------------
__global__ void __launch_bounds__(384)
filter_gemm_kernel(const float* __restrict__ x,
                   const float* __restrict__ Hp,
                   float* __restrict__ q) {
  __shared__ float As[2][16 * KP];

  const int tid   = threadIdx.x;
  const int wave  = tid >> 5;        // 0..11 -> N tile
  const int lane  = tid & 31;
  const int half  = lane >> 4;       // 0 or 1
  const int ln    = lane & 15;
  const int j0    = wave * 16;
  const int tile0 = blockIdx.x * TPB;

  const unsigned ldsBase  = (unsigned)(uintptr_t)&As[0][0];
  const unsigned bufBytes = 16u * KP * 4u;

  // Prime the pipeline: wave 0 DMAs tile 0 into buffer 0.
  if (tid < 32) {
    tdm_load_tile(ldsBase, x + (size_t)tile0 * 16 * DD);
  }

  for (int t = 0; t < TPB; ++t) {
    if (tid < 32) {
      if (t + 1 < TPB) {
        // Prefetch next tile into the other buffer, then wait for tile t
        // (TDM ops of one wave complete in order => tensorcnt<=1 means done).
        tdm_load_tile(ldsBase + (unsigned)((t + 1) & 1) * bufBytes,
                      x + (size_t)(tile0 + t + 1) * 16 * DD);
        __builtin_amdgcn_s_wait_tensorcnt(1);
      } else {
        __builtin_amdgcn_s_wait_tensorcnt(0);
      }
    }
    __syncthreads();                 // tile t visible to all waves

    const float* Asb = &As[t & 1][0];
    v8f acc = {0.f, 0.f, 0.f, 0.f, 0.f, 0.f, 0.f, 0.f};

    for (int kb = 0; kb < KP; kb += 4) {
      // A fragment (16x4 f32): VGPR0 -> K=kb+2*half, VGPR1 -> K=kb+2*half+1
      int kk = kb + 2 * half;
      v2f a, b;
      a.x = Asb[ln * KP + kk];
      a.y = Asb[ln * KP + kk + 1];
      // B fragment (4x16 f32), mirrored layout: N=ln, K=kb+2*half (+1)
      b.x = Hp[kk * NP + j0 + ln];
      b.y = Hp[(kk + 1) * NP + j0 + ln];
      acc = __builtin_amdgcn_wmma_f32_16x16x4_f32(
          /*neg_a=*/false, a, /*neg_b=*/false, b,
          /*c_mod=*/(short)0, acc, /*reuse_a=*/false, /*reuse_b=*/false);
    }

    // C/D layout: VGPR v, lanes 0-15 -> M=v, lanes 16-31 -> M=v+8, N=ln
    const int r0 = (tile0 + t) * 16;
#pragma unroll
    for (int v = 0; v < 8; ++v) {
      q[(r0 + v + 8 * half) * NP + j0 + ln] = acc[v];
    }
    __syncthreads();                 // all reads of buf[t&1] done before the
                                     // DMA for tile t+2 overwrites it
  }
}

// ---------------------------------------------------------------------------
// Kernel 3: backprojection. One block = 256 threads = 256 pixels of one image.
// Per-block f64 cos/sin table in LDS; f64 detector coordinate math (matches
// the numpy float64 table build), f32 lerp accumulation. q is L2-resident.
// ---------------------------------------------------------------------------
__global__ void __launch_bounds__(256)
backproj_kernel(const float* __restrict__ q, float* __restrict__ out) {
  __shared__ double cs[AA];
  __shared__ double sn[AA];

  const int tid = threadIdx.x;
  for (int a = tid; a < AA; a += 256) {
    double th = ((double)a + 0.5) * (PI_D / (double)AA);
    cs[a] = cos(th);
    sn[a] = sin(th);
  }
  __syncthreads();

  const int b = blockIdx.x >> 6;                   // image index
  const int p = ((blockIdx.x & 63) << 8) + tid;    // pixel index 0..16383
  const int ix = p >> 7;
  const int iy = p & 127;

  const double RHO  = 28.284271247461902;
  const double DC   = 2.0 * RHO / (double)DD;
  const double CELL = 0.3125;                      // 40/128
  const double X = -20.0 + ((double)ix + 0.5) * CELL;
  const double Y = -20.0 + ((double)iy + 0.5) * CELL;
  const double koff = RHO - 0.5 * DC;
  const double invDC = 1.0 / DC;

  const float* qb = q + (size_t)b * AA * NP;
  float accv = 0.0f;

  for (int a = 0; a < AA; ++a) {
    const float* row = qb + a * NP;
    if (a + 1 < AA) __builtin_prefetch(qb + (a + 1) * NP, 0, 3);
    double t = cs[a] * X + sn[a] * Y;
    double k = (t + koff) * invDC;
    int k0 = (int)floor(k);
    k0 = k0 < 0 ? 0 : (k0 > DD - 2 ? DD - 2 : k0);
    double wd = k - (double)k0;
    wd = wd < 0.0 ? 0.0 : (wd > 1.0 ? 1.0 : wd);
    float w = (float)wd;
    float g0 = row[k0];
    float g1 = row[k0 + 1];
    accv += (1.0f - w) * g0 + w * g1;
  }

  out[(size_t)b * (NN * NN) + p] = accv * (float)(PI_D / (double)AA);
}

// ---------------------------------------------------------------------------
extern "C" void kernel_launch(void* const* d_in, const int* in_sizes, int n_in,
                              void* d_out, int out_size, void* d_ws, size_t ws_size,
                              hipStream_t stream) {
  (void)in_sizes; (void)n_in; (void)out_size; (void)ws_size;
  const float* x = (const float*)d_in[0];
  float* out = (float*)d_out;

  float* Hp = (float*)d_ws;                              // 184*192*4 = 141 KB
  float* q  = (float*)((char*)d_ws + (size_t)(1 << 18)); // +256 KB, 56 MB

  build_h_kernel<<<dim3((KP * NP + 255) / 256), dim3(256), 0, stream>>>(Hp);
  filter_gemm_kernel<<<dim3(MM / (16 * TPB)), dim3(384), 0, stream>>>(x, Hp, q);
  backproj_kernel<<<dim3(BATCH * 64), dim3(256), 0, stream>>>(q, out);
}